// Repro_11879879541127
// MI455X (gfx1250) — compile-verified
//
#include <hip/hip_runtime.h>

// Box-filter (1024x1024 constant kernel, reflect pad) == const * separable
// sliding-window sum. Implemented O(HW) via prefix sums:
//   pass1a/1b/1c : column prefix sums C[ch][r][c] (segmented for parallelism)
//   rowKernel    : per (ch, row i) build A[i][*] from <=6 C rows, prefix-scan
//                  the 1024-wide row with V_WMMA_F32_16X16X4_F32 (M @ upper-
//                  triangular-ones, 16 chained WMMAs), then emit outputs.
// Memory-bound: ~125 MB DRAM traffic -> ~5-6 us at 23.3 TB/s; C (25 MB) is
// L2-resident (192 MB L2) so the 6x gather re-reads are cheap.

typedef __attribute__((ext_vector_type(2))) float v2f;
typedef __attribute__((ext_vector_type(8))) float v8f;

#define KCONST 2.5652997311834374e-21f
#define CH 6          // 2*3 channels
#define HW 1024

__device__ __forceinline__ int imin(int a, int b) { return a < b ? a : b; }
__device__ __forceinline__ int imax(int a, int b) { return a > b ? a : b; }
__device__ __forceinline__ v2f ld2(const float* p, int x) { return *(const v2f*)(p + x); }

// ---------------- pass 1a: per-128-row segment sums (per column) -------------
__global__ __launch_bounds__(256) void colSegSum(const float* __restrict__ x,
                                                 float* __restrict__ seg) {
    int b = blockIdx.x;                 // 6 ch * 8 seg * 4 colblk = 192
    int ch = b >> 5, rem = b & 31, sg = rem >> 2, cb = rem & 3;
    int col = cb * 256 + threadIdx.x;
    const float* xp = x + (ch << 20) + (sg * 128) * HW + col;
    float s = 0.f;
    #pragma unroll 4
    for (int r = 0; r < 128; ++r) s += xp[r << 10];
    seg[(ch * 8 + sg) * HW + col] = s;
}

// ---------------- pass 1b: exclusive scan of 8 segment sums per column -------
__global__ __launch_bounds__(256) void segScanK(float* __restrict__ seg) {
    int b = blockIdx.x;                 // 6 ch * 4 colblk = 24
    int ch = b >> 2, cb = b & 3;
    int col = cb * 256 + threadIdx.x;
    float run = 0.f;
    #pragma unroll
    for (int s = 0; s < 8; ++s) {
        int idx = (ch * 8 + s) * HW + col;
        float t = seg[idx];
        seg[idx] = run;
        run += t;
    }
}

// ---------------- pass 1c: inclusive column prefix sums C --------------------
__global__ __launch_bounds__(256) void colScan(const float* __restrict__ x,
                                               const float* __restrict__ seg,
                                               float* __restrict__ Cb) {
    int b = blockIdx.x;                 // 192 blocks
    int ch = b >> 5, rem = b & 31, sg = rem >> 2, cb = rem & 3;
    int col = cb * 256 + threadIdx.x;
    const float* xp = x + (ch << 20) + (sg * 128) * HW + col;
    float* cp = Cb + (ch << 20) + (sg * 128) * HW + col;
    float run = seg[(ch * 8 + sg) * HW + col];
    for (int r = 0; r < 128; ++r) { run += xp[r << 10]; cp[r << 10] = run; }
}

__device__ __forceinline__ float waveInclScan(float v, int lane) {
    #pragma unroll
    for (int d = 1; d < 32; d <<= 1) {
        float t = __shfl_up(v, d, 32);
        if (lane >= d) v += t;
    }
    return v;
}

// ---------------- pass 2: one wave per (channel, output row) -----------------
__global__ __launch_bounds__(32) void rowKernel(const float* __restrict__ Cb,
                                                float* __restrict__ out) {
    __shared__ float P[1024];   // inclusive row prefixes (intra-segment part)
    __shared__ float Off[64];   // per-16-segment carry offsets

    const int lane = threadIdx.x;
    const int ch = blockIdx.x >> 10;
    const int i  = blockIdx.x & 1023;
    const float* Cch = Cb + (ch << 20);

    // Row-window combination coefficients (uniform per block).
    const int r2lo = imax(i - 511, 0);
    const int r2hi = imin(i + 512, 1023) + 1;
    const float* rowHi   = Cch + (r2hi - 1) * HW;
    const float* rowLo   = Cch + imax(r2lo - 1, 0) * HW;
    const float  sLo     = (r2lo > 0) ? 1.f : 0.f;
    const float  s1      = (i <= 510) ? 1.f : 0.f;      // top reflection
    const float* rowT    = Cch + imax(511 - i, 0) * HW; // Cf(512-i)
    const float* row1    = Cch;                         // Cf(1)
    const float  s3      = (i >= 512) ? 1.f : 0.f;      // bottom reflection
    const float* row1023 = Cch + 1022 * HW;             // Cf(1023)
    const float* rowB    = Cch + imin(1533 - i, 1023) * HW; // Cf(1534-i)

    const int m  = lane & 15;         // A row / B,C,D column within tile
    const int hl = (lane >> 4) & 1;   // lane half selects K/M halves
    const int hl8 = hl << 3;

    // Intra-segment inclusive prefix via M(16x16) @ U(16x16) per tile,
    // chained as 4x V_WMMA_F32_16X16X4_F32 (K=4 each).
    #pragma unroll
    for (int t = 0; t < 4; ++t) {
        v8f acc = {};
        #pragma unroll
        for (int k = 0; k < 4; ++k) {
            const int kk = 4 * k + 2 * hl;              // K indices {kk, kk+1}
            const int x  = 256 * t + 16 * m + kk;       // element of A-row
            // A[i][x], A[i][x+1] built from <=6 C rows (branchless, uniform)
            v2f a = ld2(rowHi, x)
                  - sLo * ld2(rowLo, x)
                  + s1 * (ld2(rowT, x) - ld2(row1, x))
                  + s3 * (ld2(row1023, x) - ld2(rowB, x));
            // B = U[kk..kk+1][q], q = lane%16; U[K][q] = (K <= q)
            v2f b;
            b.x = (kk     <= m) ? 1.f : 0.f;
            b.y = (kk + 1 <= m) ? 1.f : 0.f;
            acc = __builtin_amdgcn_wmma_f32_16x16x4_f32(
                false, a, false, b, (short)0, acc, false, false);
        }
        // D layout: lane%16 = col q, VGPR v -> row v + 8*half; seg = 16t+row
        const int srow = 16 * t + hl8;
        #pragma unroll
        for (int v = 0; v < 8; ++v)
            P[(srow + v) * 16 + m] = acc[v];
    }
    __syncthreads();

    // Cross-segment carries: shuffle-scan the 64 segment totals.
    float t0 = P[lane * 16 + 15];
    float t1 = P[(lane + 32) * 16 + 15];
    float sc0 = waveInclScan(t0, lane);
    float tot0 = __shfl(sc0, 31, 32);
    float sc1 = waveInclScan(t1, lane);
    Off[lane]      = sc0 - t0;          // exclusive offsets, segments 0..31
    Off[lane + 32] = tot0 + sc1 - t1;   // segments 32..63
    __syncthreads();

    // Bpre(k) = sum_{c<k} A[i][c]
    auto Bpre = [&](int k) -> float {
        return (k == 0) ? 0.f : (P[k - 1] + Off[(k - 1) >> 4]);
    };
    const float bp1 = Bpre(1), bp1023 = Bpre(1023);
    float* op = out + (ch << 20) + i * HW;
    for (int j = lane; j < HW; j += 32) {
        float v = Bpre(imin(j + 512, 1023) + 1) - Bpre(imax(j - 511, 0));
        if (j <= 510) v += Bpre(512 - j) - bp1;
        if (j >= 512) v += bp1023 - Bpre(1534 - j);
        op[j] = KCONST * v;
    }
}

extern "C" void kernel_launch(void* const* d_in, const int* in_sizes, int n_in,
                              void* d_out, int out_size, void* d_ws, size_t ws_size,
                              hipStream_t stream) {
    (void)in_sizes; (void)n_in; (void)out_size; (void)ws_size;
    // inputs: [0]=H(int), [1]=W(int), [2]=x(fp32 2*3*1024*1024), [3]=Kh, [4]=Kw
    const float* x = (const float*)d_in[2];
    float* out = (float*)d_out;

    float* Cbuf = (float*)d_ws;                 // 6*1024*1024 floats (25.2 MB)
    float* seg  = Cbuf + CH * HW * HW;          // 6*8*1024 floats (192 KB)

    colSegSum<<<192, 256, 0, stream>>>(x, seg);
    segScanK <<< 24, 256, 0, stream>>>(seg);
    colScan  <<<192, 256, 0, stream>>>(x, seg, Cbuf);
    rowKernel<<<CH * HW, 32, 0, stream>>>(Cbuf, out);
}